// NumDecoder_23630910063056
// MI455X (gfx1250) — compile-verified
//
#include <hip/hip_runtime.h>
#include <hip/hip_bf16.h>
#include <math.h>

// Problem constants (match reference)
#define BB 512
#define SS 140
#define HH 512
#define EE 256
#define VV 1000
#define TT 21
#define NDAYS 7
#define DLEN 20
#define SOS_TOK 2

typedef __attribute__((ext_vector_type(2))) float v2f;
typedef __attribute__((ext_vector_type(8))) float v8f;

// ---------------------------------------------------------------------------
// LDS-tiled WMMA f32 GEMM:  C[M,N] = A[M,K] @ W[N,K]^T + bias, optional tanh.
// Block = 256 threads (8 waves, arranged 4 row-waves x 2 col-waves).
// Block tile = 64 rows x 128 cols; K chunked by 32.
// LDS row stride 36 floats: 16B-aligned b128 stores + conflict-free b64 reads
// (36*r mod 64 distinct for r=0..15).
// Fragment layout per ISA 7.12.2 (16x4 A / 4x16 B):
//   lanes 0-15 hold K=k,k+1 ; lanes 16-31 hold K=k+2,k+3.
// ---------------------------------------------------------------------------
#define KC 32
#define LDSTR 36

__global__ __launch_bounds__(256) void gemm_wmma_tn(
    const float* __restrict__ A, const float* __restrict__ W,
    const float* __restrict__ bias, float* __restrict__ C,
    int M, int N, int K, int act) {
  __shared__ float sA[64 * LDSTR];
  __shared__ float sW[128 * LDSTR];

  const int tid  = threadIdx.x;
  const int wave = tid >> 5;
  const int lane = tid & 31;
  const int l15  = lane & 15;
  const int hi   = lane >> 4;            // 0: K=k,k+1  1: K=k+2,k+3
  const int wr16 = (wave & 3) * 16;      // wave row offset within block tile
  const int wc64 = (wave >> 2) * 64;     // wave col offset within block tile
  const int rowblk = blockIdx.y * 64;
  const int col0   = blockIdx.x * 128;

  v8f acc[4] = {};

  // tile-load index mapping: 8 float4s per 32-float row
  const int lr = tid >> 3;               // 0..31
  const int lc = (tid & 7) * 4;          // 0,4,..,28

  for (int k0 = 0; k0 < K; k0 += KC) {
    // ---- stage A tile: 64 rows x 32 floats (2 float4 per thread) ----
#pragma unroll
    for (int i = 0; i < 2; ++i) {
      int r = lr + i * 32;
      *(float4*)&sA[r * LDSTR + lc] =
          *(const float4*)&A[(size_t)(rowblk + r) * K + k0 + lc];
    }
    // ---- stage W tile: 128 rows x 32 floats (4 float4 per thread) ----
#pragma unroll
    for (int i = 0; i < 4; ++i) {
      int r = lr + i * 32;
      int wrow = col0 + r;
      if (wrow > N - 1) wrow = N - 1;    // clamp; stores are guarded
      *(float4*)&sW[r * LDSTR + lc] =
          *(const float4*)&W[(size_t)wrow * K + k0 + lc];
    }
    // prefetch hint for next K chunk (global_prefetch_b8)
    if (k0 + KC < K) {
      __builtin_prefetch(&A[(size_t)(rowblk + lr) * K + k0 + KC + lc], 0, 0);
      int wrow = col0 + lr;
      if (wrow > N - 1) wrow = N - 1;
      __builtin_prefetch(&W[(size_t)wrow * K + k0 + KC + lc], 0, 0);
    }
    __syncthreads();

    // ---- compute: 8 k-steps x 4 WMMAs per wave ----
#pragma unroll
    for (int k = 0; k < KC; k += 4) {
      v2f a = *(const v2f*)&sA[(wr16 + l15) * LDSTR + k + 2 * hi];
#pragma unroll
      for (int j = 0; j < 4; ++j) {
        v2f b = *(const v2f*)&sW[(wc64 + j * 16 + l15) * LDSTR + k + 2 * hi];
        acc[j] = __builtin_amdgcn_wmma_f32_16x16x4_f32(
            false, a, false, b, (short)0, acc[j], false, false);
      }
    }
    __syncthreads();
  }

  // ---- epilogue: C/D layout VGPR v -> (row + v + 8*hi, col) ----
#pragma unroll
  for (int j = 0; j < 4; ++j) {
    int c = col0 + wc64 + j * 16 + l15;
    if (c < N) {
      float bv = bias ? bias[c] : 0.0f;
#pragma unroll
      for (int v = 0; v < 8; ++v) {
        int r = rowblk + wr16 + hi * 8 + v;
        float o = acc[j][v] + bv;
        if (act == 1) o = tanhf(o);
        C[(size_t)r * N + c] = o;
      }
    }
  }
}

// ---------------------------------------------------------------------------
// 32x32 tiled transpose: dst[j,i] = src[i,j]  (square n x n)
// ---------------------------------------------------------------------------
__global__ __launch_bounds__(256) void transpose_k(
    const float* __restrict__ src, float* __restrict__ dst, int n) {
  __shared__ float tile[32][33];
  const int bx = blockIdx.x * 32, by = blockIdx.y * 32;
  const int tx = threadIdx.x & 31, ty = threadIdx.x >> 5;
#pragma unroll
  for (int i = 0; i < 32; i += 8)
    tile[ty + i][tx] = src[(size_t)(by + ty + i) * n + bx + tx];
  __syncthreads();
#pragma unroll
  for (int i = 0; i < 32; i += 8)
    dst[(size_t)(bx + ty + i) * n + by + tx] = tile[tx][ty + i];
}

// ---------------------------------------------------------------------------
// Day-level attention, fused: scores (U . enc), mask, softmax over 20, context.
// One 256-thread block per (b, day).
// ---------------------------------------------------------------------------
__global__ __launch_bounds__(256) void attn_day_k(
    const float* __restrict__ U, const float* __restrict__ enc,
    const int* __restrict__ numpairs, float* __restrict__ week) {
  const int bd = blockIdx.x;
  const int b = bd / NDAYS, d = bd % NDAYS;
  __shared__ float sU[HH];
  __shared__ float s_sc[DLEN];
  __shared__ float s_wd[DLEN];
  const int tid = threadIdx.x;
  sU[tid]       = U[(size_t)b * HH + tid];
  sU[tid + 256] = U[(size_t)b * HH + tid + 256];
  __syncthreads();

  const int wave = tid >> 5, lane = tid & 31;
  const float* encbd = enc + ((size_t)b * SS + d * DLEN) * HH;

  for (int s = wave; s < DLEN; s += 8) {
    const float* row = encbd + (size_t)s * HH;
    float acc = 0.0f;
    for (int i = lane * 4; i < HH; i += 128) {
      float4 u = *(const float4*)(sU + i);
      float4 r = *(const float4*)(row + i);
      acc += u.x * r.x + u.y * r.y + u.z * r.z + u.w * r.w;
    }
    for (int off = 16; off; off >>= 1) acc += __shfl_xor(acc, off, 32);
    if (lane == 0) {
      bool valid = numpairs[(size_t)b * SS + d * DLEN + s] != 0;
      s_sc[s] = valid ? acc : -1e9f;
    }
  }
  __syncthreads();
  if (tid == 0) {
    float m = s_sc[0];
    for (int s = 1; s < DLEN; ++s) m = fmaxf(m, s_sc[s]);
    float sum = 0.0f;
    for (int s = 0; s < DLEN; ++s) { float e = expf(s_sc[s] - m); s_wd[s] = e; sum += e; }
    float inv = 1.0f / sum;
    for (int s = 0; s < DLEN; ++s) s_wd[s] *= inv;
  }
  __syncthreads();

  float a0 = 0.0f, a1 = 0.0f;
  for (int s = 0; s < DLEN; ++s) {
    float w = s_wd[s];
    a0 += w * encbd[(size_t)s * HH + tid];
    a1 += w * encbd[(size_t)s * HH + tid + 256];
  }
  week[((size_t)b * NDAYS + d) * HH + tid]       = a0;
  week[((size_t)b * NDAYS + d) * HH + tid + 256] = a1;
}

// ---------------------------------------------------------------------------
// Week-level attention over 7 day contexts + build concat = [h_new | ctx].
// One 256-thread block per b.
// ---------------------------------------------------------------------------
__global__ __launch_bounds__(256) void week_attn_k(
    const float* __restrict__ U2, const float* __restrict__ week,
    const float* __restrict__ hnew, float* __restrict__ cat) {
  const int b = blockIdx.x;
  __shared__ float sU[HH];
  __shared__ float s_sc[8];
  __shared__ float s_aw[8];
  const int tid = threadIdx.x;
  sU[tid]       = U2[(size_t)b * HH + tid];
  sU[tid + 256] = U2[(size_t)b * HH + tid + 256];
  __syncthreads();

  const int wave = tid >> 5, lane = tid & 31;
  if (wave < NDAYS) {
    const float* row = week + ((size_t)b * NDAYS + wave) * HH;
    float acc = 0.0f;
    for (int i = lane * 4; i < HH; i += 128) {
      float4 u = *(const float4*)(sU + i);
      float4 r = *(const float4*)(row + i);
      acc += u.x * r.x + u.y * r.y + u.z * r.z + u.w * r.w;
    }
    for (int off = 16; off; off >>= 1) acc += __shfl_xor(acc, off, 32);
    if (lane == 0) s_sc[wave] = acc;
  }
  __syncthreads();
  if (tid == 0) {
    float m = s_sc[0];
    for (int d = 1; d < NDAYS; ++d) m = fmaxf(m, s_sc[d]);
    float sum = 0.0f;
    for (int d = 0; d < NDAYS; ++d) { float e = expf(s_sc[d] - m); s_aw[d] = e; sum += e; }
    float inv = 1.0f / sum;
    for (int d = 0; d < NDAYS; ++d) s_aw[d] *= inv;
  }
  __syncthreads();

  float c0 = 0.0f, c1 = 0.0f;
  for (int d = 0; d < NDAYS; ++d) {
    float a = s_aw[d];
    c0 += a * week[((size_t)b * NDAYS + d) * HH + tid];
    c1 += a * week[((size_t)b * NDAYS + d) * HH + tid + 256];
  }
  float* cb = cat + (size_t)b * (2 * HH);
  cb[tid]            = hnew[(size_t)b * HH + tid];
  cb[tid + 256]      = hnew[(size_t)b * HH + tid + 256];
  cb[HH + tid]       = c0;
  cb[HH + tid + 256] = c1;
}

// ---------------------------------------------------------------------------
// GRU gate merge (PyTorch r,z,n layout).
// ---------------------------------------------------------------------------
__global__ __launch_bounds__(256) void gru_update_k(
    const float* __restrict__ gi, const float* __restrict__ gh,
    const float* __restrict__ hprev, float* __restrict__ hnew) {
  const int idx = blockIdx.x * 256 + threadIdx.x;  // b*H + h
  const int b = idx >> 9, h = idx & (HH - 1);
  const float* gib = gi + (size_t)b * (3 * HH);
  const float* ghb = gh + (size_t)b * (3 * HH);
  float ir = gib[h], iz = gib[HH + h], in_ = gib[2 * HH + h];
  float hr = ghb[h], hz = ghb[HH + h], hn = ghb[2 * HH + h];
  float r = 1.0f / (1.0f + expf(-(ir + hr)));
  float z = 1.0f / (1.0f + expf(-(iz + hz)));
  float n = tanhf(in_ + r * hn);
  hnew[idx] = (1.0f - z) * n + z * hprev[idx];
}

// ---------------------------------------------------------------------------
// Embedding gather with teacher forcing (t==0 -> SOS).
// ---------------------------------------------------------------------------
__global__ __launch_bounds__(256) void embed_k(
    const float* __restrict__ emb, const int* __restrict__ label, int t,
    float* __restrict__ x) {
  const int b = blockIdx.x, e = threadIdx.x;  // E == 256
  int tok = (t == 0) ? SOS_TOK : label[(size_t)b * (TT + 1) + (t - 1)];
  x[(size_t)b * EE + e] = emb[(size_t)tok * EE + e];
}

// ---------------------------------------------------------------------------
// Row-wise log-softmax over V=1000, writes into out[b, t, :].
// ---------------------------------------------------------------------------
__global__ __launch_bounds__(256) void logsoftmax_k(
    const float* __restrict__ logits, float* __restrict__ out, int t) {
  const int b = blockIdx.x, tid = threadIdx.x;
  __shared__ float red[8];
  __shared__ float stat[2];
  const float* row = logits + (size_t)b * VV;

  float m = -3.4e38f;
  for (int v = tid; v < VV; v += 256) m = fmaxf(m, row[v]);
  for (int off = 16; off; off >>= 1) m = fmaxf(m, __shfl_xor(m, off, 32));
  if ((tid & 31) == 0) red[tid >> 5] = m;
  __syncthreads();
  if (tid == 0) {
    float x = red[0];
    for (int i = 1; i < 8; ++i) x = fmaxf(x, red[i]);
    stat[0] = x;
  }
  __syncthreads();
  const float mx = stat[0];

  float s = 0.0f;
  for (int v = tid; v < VV; v += 256) s += expf(row[v] - mx);
  for (int off = 16; off; off >>= 1) s += __shfl_xor(s, off, 32);
  if ((tid & 31) == 0) red[tid >> 5] = s;
  __syncthreads();
  if (tid == 0) {
    float x = 0.0f;
    for (int i = 0; i < 8; ++i) x += red[i];
    stat[1] = mx + logf(x);
  }
  __syncthreads();
  const float lse = stat[1];
  float* orow = out + ((size_t)b * TT + t) * VV;
  for (int v = tid; v < VV; v += 256) orow[v] = row[v] - lse;
}

__global__ __launch_bounds__(256) void copy_k(const float* __restrict__ s,
                                              float* __restrict__ d, int n) {
  int i = blockIdx.x * 256 + threadIdx.x;
  if (i < n) d[i] = s[i];
}

__global__ __launch_bounds__(256) void lastin_k(const int* __restrict__ label,
                                                int* __restrict__ o) {
  int b = blockIdx.x * 256 + threadIdx.x;
  if (b < BB) o[b] = label[(size_t)b * (TT + 1) + (TT - 1)];
}

// ---------------------------------------------------------------------------
extern "C" void kernel_launch(void* const* d_in, const int* in_sizes, int n_in,
                              void* d_out, int out_size, void* d_ws,
                              size_t ws_size, hipStream_t stream) {
  const float* enc_hidden = (const float*)d_in[0];   // [1,B,H]
  const float* enc_out    = (const float*)d_in[1];   // [B,S,H]
  const int*   label      = (const int*)d_in[2];     // [B,T+1]
  const int*   numpairs   = (const int*)d_in[3];     // [B,S]
  const float* emb        = (const float*)d_in[4];   // [V,E]
  const float* W_ih       = (const float*)d_in[5];   // [3H,E]
  const float* W_hh       = (const float*)d_in[6];   // [3H,H]
  const float* b_ih       = (const float*)d_in[7];   // [3H]
  const float* b_hh       = (const float*)d_in[8];   // [3H]
  const float* attn_Wa    = (const float*)d_in[9];   // [H,H]
  const float* wa_W       = (const float*)d_in[10];  // [H,2H]
  const float* fc_W       = (const float*)d_in[11];  // [V,H]
  const float* fc_b       = (const float*)d_in[12];  // [V]
  float* out = (float*)d_out;

  float* ws = (float*)d_ws;
  size_t o = 0;
  float* WaT   = ws + o; o += (size_t)HH * HH;        // attn_Wa transposed
  float* hbuf0 = ws + o; o += (size_t)BB * HH;
  float* hbuf1 = ws + o; o += (size_t)BB * HH;
  float* x     = ws + o; o += (size_t)BB * EE;
  float* U     = ws + o; o += (size_t)BB * HH;
  float* week  = ws + o; o += (size_t)BB * NDAYS * HH;
  float* gi    = ws + o; o += (size_t)BB * 3 * HH;
  float* gh    = ws + o; o += (size_t)BB * 3 * HH;
  float* cat   = ws + o; o += (size_t)BB * 2 * HH;
  float* z1    = ws + o; o += (size_t)BB * HH;
  float* logit = ws + o; o += (size_t)BB * VV;

  // WaT[h,g] = Wa[g,h]  (so U = h @ Wa becomes the TN form with weight WaT)
  transpose_k<<<dim3(HH / 32, HH / 32), 256, 0, stream>>>(attn_Wa, WaT, HH);
  // h0 = encoder_hidden[0]
  copy_k<<<(BB * HH + 255) / 256, 256, 0, stream>>>(enc_hidden, hbuf0, BB * HH);

  for (int t = 0; t < TT; ++t) {
    float* hprev = (t & 1) ? hbuf1 : hbuf0;
    float* hnext = (t & 1) ? hbuf0 : hbuf1;

    // x = emb[dec_in(t)]
    embed_k<<<BB, 256, 0, stream>>>(emb, label, t, x);

    // U = h_prev @ Wa  (replaces energy_day precompute AND the per-step ew GEMM)
    gemm_wmma_tn<<<dim3(HH / 128, BB / 64), 256, 0, stream>>>(
        hprev, WaT, nullptr, U, BB, HH, HH, 0);

    // day attention -> week[B,7,H]
    attn_day_k<<<BB * NDAYS, 256, 0, stream>>>(U, enc_out, numpairs, week);

    // GRU gates: gi = x @ W_ih^T + b_ih ; gh = h_prev @ W_hh^T + b_hh
    gemm_wmma_tn<<<dim3(3 * HH / 128, BB / 64), 256, 0, stream>>>(
        x, W_ih, b_ih, gi, BB, 3 * HH, EE, 0);
    gemm_wmma_tn<<<dim3(3 * HH / 128, BB / 64), 256, 0, stream>>>(
        hprev, W_hh, b_hh, gh, BB, 3 * HH, HH, 0);
    gru_update_k<<<(BB * HH) / 256, 256, 0, stream>>>(gi, gh, hprev, hnext);

    // U2 = h_new @ Wa ; week attention -> concat [h_new | ctx]
    gemm_wmma_tn<<<dim3(HH / 128, BB / 64), 256, 0, stream>>>(
        hnext, WaT, nullptr, U, BB, HH, HH, 0);
    week_attn_k<<<BB, 256, 0, stream>>>(U, week, hnext, cat);

    // z1 = tanh(cat @ wa_W^T) ; logits = z1 @ fc_W^T + fc_b
    gemm_wmma_tn<<<dim3(HH / 128, BB / 64), 256, 0, stream>>>(
        cat, wa_W, nullptr, z1, BB, HH, 2 * HH, 1);
    gemm_wmma_tn<<<dim3((VV + 127) / 128, BB / 64), 256, 0, stream>>>(
        z1, fc_W, fc_b, logit, BB, VV, HH, 0);

    // out[:, t, :] = log_softmax(logits)
    logsoftmax_k<<<BB, 256, 0, stream>>>(logit, out, t);
  }

  // last_in = label[:, T-1] (int payload appended after [B,T,V] floats)
  lastin_k<<<(BB + 255) / 256, 256, 0, stream>>>(
      label, (int*)(out + (size_t)BB * TT * VV));
}